// SAGE_69028714381804
// MI455X (gfx1250) — compile-verified
//
#include <hip/hip_runtime.h>
#include <math.h>

#define DH 128          // feature dim (both layers)
#define LDS_STRIDE 132  // 128 + 4 pad -> spreads LDS banks for column reads

typedef __attribute__((ext_vector_type(2))) float v2f;
typedef __attribute__((ext_vector_type(8))) float v8f;

// ---------------------------------------------------------------- utilities

__global__ void sage_fill_zero(float* __restrict__ p, long long count) {
    long long i = (long long)blockIdx.x * blockDim.x + threadIdx.x;
    long long stride = (long long)gridDim.x * blockDim.x;
    for (; i < count; i += stride) p[i] = 0.0f;
}

// deg[dst[e]] += 1
__global__ void sage_degree(const long long* __restrict__ dst,
                            float* __restrict__ deg, int nE) {
    int i = blockIdx.x * blockDim.x + threadIdx.x;
    if (i < nE) atomicAdd(&deg[dst[i]], 1.0f);
}

// deg[i] <- 1 / max(deg[i], 1)
__global__ void sage_invdeg(float* __restrict__ deg, int n) {
    int i = blockIdx.x * blockDim.x + threadIdx.x;
    if (i < n) deg[i] = 1.0f / fmaxf(deg[i], 1.0f);
}

// Pack W[128][128] into WMMA B-fragment pair order:
//   Wp[p*256 + col*2 + (k&1)] = W[k][col],  p = k>>1
// so a lane's (b.x, b.y) = (W[k][c], W[k+1][c]) is one contiguous float2.
__global__ void sage_pack_w(const float* __restrict__ W, float* __restrict__ Wp) {
    int i = blockIdx.x * blockDim.x + threadIdx.x;   // 0..16383
    if (i < DH * DH) {
        int col = i & (DH - 1);
        int k   = i >> 7;
        Wp[(k >> 1) * (2 * DH) + col * 2 + (k & 1)] = W[i];
    }
}

// ------------------------------------------------- edge scatter (wave/edge)
// agg[dst[e], :] += h[src[e], :]   (128 floats = 32 lanes x float4)
__global__ void sage_scatter(const float* __restrict__ h,
                             const long long* __restrict__ src,
                             const long long* __restrict__ dst,
                             float* __restrict__ agg, int nE) {
    int wid  = blockIdx.x * (blockDim.x >> 5) + (threadIdx.x >> 5);
    int lane = threadIdx.x & 31;
    if (wid >= nE) return;
    long long s = src[wid];
    long long d = dst[wid];
    const float4* hp = (const float4*)(h + s * (long long)DH);
    float4 v = hp[lane];                       // one 512B row per wave
    float* ap = agg + d * (long long)DH + lane * 4;
    atomicAdd(ap + 0, v.x);
    atomicAdd(ap + 1, v.y);
    atomicAdd(ap + 2, v.z);
    atomicAdd(ap + 3, v.w);
}

// ------------------------------------- fused SAGE layer GEMM (fp32 WMMA)
// Out[i,:] = relu( Hs[i,:] @ Ws  +  (Agg[i,:] * invdeg[i]) @ Wn  + bias )
// Block = 256 threads = 8 waves, owns a 32-row panel. Wave w owns columns
// [16w,16w+16) and two 16x16 accumulators (rows 0-15 / 16-31 of the panel),
// so each B fragment (packed float2, single global_load_b64) feeds 2 WMMAs.
// Bias is folded into c0's init, added to c1 at store (all 8 D VGPRs of a lane share one
// output column).
__global__ __launch_bounds__(256)
void sage_gemm_wmma(const float* __restrict__ Hs,
                    const float* __restrict__ Agg,
                    const float* __restrict__ invdeg,
                    const float* __restrict__ WsP,   // packed pair-order
                    const float* __restrict__ WnP,   // packed pair-order
                    const float* __restrict__ bias,
                    float* __restrict__ Out, int n) {
    __shared__ float As[32][LDS_STRIDE];   // self features
    __shared__ float Aa[32][LDS_STRIDE];   // pre-scaled neighbor aggregate

    const int t    = threadIdx.x;
    const int row0 = blockIdx.x * 32;

    // ---- cooperative stage of the 32x128 A panels into LDS --------------
    {
        int lr = t >> 3;              // 0..31 : panel row
        int lc = (t & 7) * 16;        // 0..112: col start (16 floats/thread)
        long long gr = (long long)row0 + lr;
        if (gr < n) {
            const float4* ph = (const float4*)(Hs + gr * (long long)DH + lc);
            const float4* pa = (const float4*)(Agg + gr * (long long)DH + lc);
            float sc = invdeg[gr];
            #pragma unroll
            for (int q = 0; q < 4; ++q) {
                float4 hv = ph[q];
                As[lr][lc + 4 * q + 0] = hv.x;
                As[lr][lc + 4 * q + 1] = hv.y;
                As[lr][lc + 4 * q + 2] = hv.z;
                As[lr][lc + 4 * q + 3] = hv.w;
                float4 av = pa[q];
                Aa[lr][lc + 4 * q + 0] = av.x * sc;
                Aa[lr][lc + 4 * q + 1] = av.y * sc;
                Aa[lr][lc + 4 * q + 2] = av.z * sc;
                Aa[lr][lc + 4 * q + 3] = av.w * sc;
            }
        } else {
            #pragma unroll
            for (int j = 0; j < 16; ++j) { As[lr][lc + j] = 0.0f; Aa[lr][lc + j] = 0.0f; }
        }
    }
    __syncthreads();

    // ---- WMMA accumulation ---------------------------------------------
    const int lane = t & 31;
    const int w    = t >> 5;            // wave id -> column tile
    const int col0 = w * 16;
    const int m    = lane & 15;         // A row within tile / D column
    const int koff = (lane >> 4) * 2;   // lanes 0-15: K+0,K+1 ; 16-31: K+2,K+3
    const int ccol = col0 + m;          // this lane's output column
    const int phl  = lane >> 4;         // K-pair half select (0 or 1)

    const float bb = bias[ccol];
    v8f c0, c1;
    #pragma unroll
    for (int i = 0; i < 8; ++i) { c0[i] = bb; c1[i] = 0.0f; }

    const float2* WsF = (const float2*)WsP;   // float2 index = p*128 + col
    const float2* WnF = (const float2*)WnP;

    #pragma unroll 4
    for (int k0 = 0; k0 < DH; k0 += 4) {
        int p = (k0 >> 1) + phl;              // K-pair index (k0+koff)/2
        float2 bv = WsF[p * DH + ccol];       // single b64, fully coalesced
        v2f a0, a1, b;
        b.x  = bv.x;                b.y  = bv.y;
        a0.x = As[m][k0 + koff];    a0.y = As[m][k0 + koff + 1];
        a1.x = As[m + 16][k0 + koff]; a1.y = As[m + 16][k0 + koff + 1];
        c0 = __builtin_amdgcn_wmma_f32_16x16x4_f32(
                 false, a0, false, b, (short)0, c0, false, false);
        c1 = __builtin_amdgcn_wmma_f32_16x16x4_f32(
                 false, a1, false, b, (short)0, c1, false, false);
    }
    #pragma unroll 4
    for (int k0 = 0; k0 < DH; k0 += 4) {
        int p = (k0 >> 1) + phl;
        float2 bv = WnF[p * DH + ccol];
        v2f a0, a1, b;
        b.x  = bv.x;                b.y  = bv.y;
        a0.x = Aa[m][k0 + koff];    a0.y = Aa[m][k0 + koff + 1];
        a1.x = Aa[m + 16][k0 + koff]; a1.y = Aa[m + 16][k0 + koff + 1];
        c0 = __builtin_amdgcn_wmma_f32_16x16x4_f32(
                 false, a0, false, b, (short)0, c0, false, false);
        c1 = __builtin_amdgcn_wmma_f32_16x16x4_f32(
                 false, a1, false, b, (short)0, c1, false, false);
    }

    // ---- ReLU + store ---------------------------------------------------
    // D layout: VGPR i -> row (i + 8*(lane>=16)), column = lane&15 (+col0)
    const int rb0 = row0 + ((lane >> 4) * 8);        // c0 rows
    #pragma unroll
    for (int i = 0; i < 8; ++i) {
        long long r = (long long)rb0 + i;
        if (r < n) {
            float v = c0[i];
            Out[r * (long long)DH + ccol] = v > 0.0f ? v : 0.0f;
        }
    }
    const int rb1 = row0 + 16 + ((lane >> 4) * 8);   // c1 rows
    #pragma unroll
    for (int i = 0; i < 8; ++i) {
        long long r = (long long)rb1 + i;
        if (r < n) {
            float v = c1[i] + bb;                    // bias went into c0 only
            Out[r * (long long)DH + ccol] = v > 0.0f ? v : 0.0f;
        }
    }
}

// ----------------------------------------------- row L2 normalize (in place)
__global__ void sage_l2norm(float* __restrict__ out, int n) {
    int wid  = blockIdx.x * (blockDim.x >> 5) + (threadIdx.x >> 5);
    int lane = threadIdx.x & 31;
    if (wid >= n) return;
    float4* p = (float4*)(out + (long long)wid * DH);
    float4 v  = p[lane];
    float ss = v.x * v.x + v.y * v.y + v.z * v.z + v.w * v.w;
    #pragma unroll
    for (int off = 16; off > 0; off >>= 1) ss += __shfl_xor(ss, off, 32);
    float norm = sqrtf(ss);
    float sc   = 1.0f / fmaxf(norm, 1e-12f);
    v.x *= sc; v.y *= sc; v.z *= sc; v.w *= sc;
    p[lane] = v;
}

// --------------------------------------------------------------- launcher
extern "C" void kernel_launch(void* const* d_in, const int* in_sizes, int n_in,
                              void* d_out, int out_size, void* d_ws, size_t ws_size,
                              hipStream_t stream) {
    const float*     x   = (const float*)d_in[0];
    const long long* src = (const long long*)d_in[1];   // int64 per reference
    const long long* dst = (const long long*)d_in[2];
    const float*     Ws0 = (const float*)d_in[3];
    const float*     Wn0 = (const float*)d_in[4];
    const float*     b0  = (const float*)d_in[5];
    const float*     Ws1 = (const float*)d_in[6];
    const float*     Wn1 = (const float*)d_in[7];
    const float*     b1  = (const float*)d_in[8];

    const int n  = in_sizes[0] / DH;   // 100000 nodes
    const int nE = in_sizes[1];        // 1.6M edges
    float* out = (float*)d_out;

    // workspace: deg [n] | agg [n*128] | h1 [n*128] | 4 packed W [16384 each]
    float* deg  = (float*)d_ws;
    float* agg  = deg + n;
    float* h1   = agg + (size_t)n * DH;
    float* Wp0s = h1 + (size_t)n * DH;
    float* Wp0n = Wp0s + DH * DH;
    float* Wp1s = Wp0n + DH * DH;
    float* Wp1n = Wp1s + DH * DH;

    const long long aggCount = (long long)n * DH;
    const int wBlocks = (DH * DH + 255) / 256;   // 64

    // pack weights into B-fragment pair order (once per launch, tiny)
    sage_pack_w<<<wBlocks, 256, 0, stream>>>(Ws0, Wp0s);
    sage_pack_w<<<wBlocks, 256, 0, stream>>>(Wn0, Wp0n);
    sage_pack_w<<<wBlocks, 256, 0, stream>>>(Ws1, Wp1s);
    sage_pack_w<<<wBlocks, 256, 0, stream>>>(Wn1, Wp1n);

    // degrees -> inverse degrees
    sage_fill_zero<<<1024, 256, 0, stream>>>(deg, n);
    sage_degree<<<(nE + 255) / 256, 256, 0, stream>>>(dst, deg, nE);
    sage_invdeg<<<(n + 255) / 256, 256, 0, stream>>>(deg, n);

    // ---- layer 0: h1 = relu(x@Ws0 + mean_agg(x)@Wn0 + b0)
    sage_fill_zero<<<4096, 256, 0, stream>>>(agg, aggCount);
    sage_scatter<<<(nE + 7) / 8, 256, 0, stream>>>(x, src, dst, agg, nE);
    sage_gemm_wmma<<<(n + 31) / 32, 256, 0, stream>>>(x, agg, deg, Wp0s, Wp0n, b0, h1, n);

    // ---- layer 1: out = relu(h1@Ws1 + mean_agg(h1)@Wn1 + b1)
    sage_fill_zero<<<4096, 256, 0, stream>>>(agg, aggCount);
    sage_scatter<<<(nE + 7) / 8, 256, 0, stream>>>(h1, src, dst, agg, nE);
    sage_gemm_wmma<<<(n + 31) / 32, 256, 0, stream>>>(h1, agg, deg, Wp1s, Wp1n, b1, out, n);

    // ---- row-wise L2 normalize in place
    sage_l2norm<<<(n + 7) / 8, 256, 0, stream>>>(out, n);
}